// WeightedGCN_BioFeatures_46626164965916
// MI455X (gfx1250) — compile-verified
//
#include <hip/hip_runtime.h>

#define HD 128

typedef __attribute__((ext_vector_type(2))) float v2f;
typedef __attribute__((ext_vector_type(8))) float v8f;

__device__ __forceinline__ float atomic_add_f32(float* p, float v) {
  return __hip_atomic_fetch_add(p, v, __ATOMIC_RELAXED, __HIP_MEMORY_SCOPE_AGENT);
}

// ---------------- LayerNorm stats over a 128-thread block (4 wave32) --------
__device__ __forceinline__ void block128_stats(float v, int c, float& mu, float& rstd) {
  __shared__ float ls[4], ls2[4];
  float s = v, s2 = v * v;
#pragma unroll
  for (int off = 16; off > 0; off >>= 1) {
    s  += __shfl_down(s, off, 32);
    s2 += __shfl_down(s2, off, 32);
  }
  if ((c & 31) == 0) { ls[c >> 5] = s; ls2[c >> 5] = s2; }
  __syncthreads();
  float ts  = ls[0] + ls[1] + ls[2] + ls[3];
  float ts2 = ls2[0] + ls2[1] + ls2[2] + ls2[3];
  mu = ts * (1.0f / HD);
  float var = ts2 * (1.0f / HD) - mu * mu;
  rstd = rsqrtf(var + 1e-5f);
}

// ---------------- input projection: x[N,8] @ W_in + b -> LN -> ReLU ---------
__global__ void __launch_bounds__(128)
k_inproj(const float* __restrict__ x, const float* __restrict__ Win,
         const float* __restrict__ bin, const float* __restrict__ g,
         const float* __restrict__ b, float* __restrict__ h) {
  int node = blockIdx.x, c = threadIdx.x;
  __shared__ float xr[8];
  if (c < 8) xr[c] = x[node * 8 + c];
  __syncthreads();
  float t = bin[c];
#pragma unroll
  for (int k = 0; k < 8; ++k) t = fmaf(xr[k], Win[k * HD + c], t);
  float mu, rstd;
  block128_stats(t, c, mu, rstd);
  float y = (t - mu) * rstd * g[c] + b[c];
  h[node * HD + c] = fmaxf(y, 0.0f);
}

// ---------------- first LayerNorm (no residual input) -----------------------
__global__ void __launch_bounds__(128)
k_ln(const float* __restrict__ in, float* __restrict__ out,
     const float* __restrict__ g, const float* __restrict__ b) {
  int node = blockIdx.x, c = threadIdx.x;
  float v = in[node * HD + c];
  float mu, rstd;
  block128_stats(v, c, mu, rstd);
  out[node * HD + c] = (v - mu) * rstd * g[c] + b[c];
}

// ---- fused: h = relu(conv_out)+h ; z = LN(h) (next layer's input) ----------
__global__ void __launch_bounds__(128)
k_resid_ln(float* __restrict__ zacc, float* __restrict__ h,
           const float* __restrict__ g, const float* __restrict__ b) {
  int node = blockIdx.x, c = threadIdx.x;
  size_t idx = (size_t)node * HD + c;
  float hn = fmaxf(zacc[idx], 0.0f) + h[idx];
  h[idx] = hn;
  float mu, rstd;
  block128_stats(hn, c, mu, rstd);
  zacc[idx] = (hn - mu) * rstd * g[c] + b[c];
}

// ---------------- FP32 WMMA GEMM + fused self-loop/bias epilogue ------------
// Y[N,128] = Z[N,128] @ W[128,128];  A = Y * dinv(row)^2 + bias(col)
// Block = 256 threads (8 wave32), 32 rows per block; A may alias Z (each
// block reads only its own 32 Z-rows into LDS before writing them).
__global__ void __launch_bounds__(256)
k_gemm(const float* __restrict__ Z, const float* __restrict__ W,
       float* __restrict__ Y, float* __restrict__ A,
       const float* __restrict__ dinv, const float* __restrict__ bias) {
  __shared__ float sZ[32 * 132];  // padded stride 132 to dodge bank conflicts
  int tid = threadIdx.x;
  int row0 = blockIdx.x * 32;
  for (int i = tid; i < 32 * HD; i += 256) {
    int r = i >> 7, c = i & (HD - 1);
    sZ[r * 132 + c] = Z[(row0 + r) * HD + c];
  }
  __syncthreads();
  int lane = tid & 31;
  int wave = tid >> 5;
  int mt = wave >> 2;          // 0..1
  int nt = (wave & 3) * 32;    // column base, two 16-wide tiles
  int l15 = lane & 15;
  int hi = lane >> 4;          // half-wave: K pair select for A/B frags
  const float* zrow = sZ + (mt * 16 + l15) * 132;
  const float* wb0 = W + nt + l15;
  const float* wb1 = W + nt + 16 + l15;
  v8f acc0 = {0.f, 0.f, 0.f, 0.f, 0.f, 0.f, 0.f, 0.f};
  v8f acc1 = {0.f, 0.f, 0.f, 0.f, 0.f, 0.f, 0.f, 0.f};
  for (int k = 0; k < HD; k += 4) {
    v2f a, b0, b1;
    // A 16x4: lanes 0-15 hold K=k,k+1; lanes 16-31 hold K=k+2,k+3 (ISA layout)
    a[0] = zrow[k + 2 * hi];
    a[1] = zrow[k + 2 * hi + 1];
    // B 4x16: mirrored layout, row-major W (K x N)
    b0[0] = wb0[(k + 2 * hi) * HD];
    b0[1] = wb0[(k + 2 * hi + 1) * HD];
    b1[0] = wb1[(k + 2 * hi) * HD];
    b1[1] = wb1[(k + 2 * hi + 1) * HD];
    acc0 = __builtin_amdgcn_wmma_f32_16x16x4_f32(false, a, false, b0, (short)0,
                                                 acc0, false, false);
    acc1 = __builtin_amdgcn_wmma_f32_16x16x4_f32(false, a, false, b1, (short)0,
                                                 acc1, false, false);
  }
  // C/D layout: VGPR r -> row r (lanes 0-15) / row r+8 (lanes 16-31)
  int rowb = row0 + mt * 16 + hi * 8;
  int col = nt + l15;
  float bc0 = bias[col], bc1 = bias[col + 16];
#pragma unroll
  for (int r = 0; r < 8; ++r) {
    int row = rowb + r;
    float dv = dinv[row];
    float dd = dv * dv;
    float y0 = acc0[r], y1 = acc1[r];
    Y[row * HD + col]      = y0;
    Y[row * HD + col + 16] = y1;
    A[row * HD + col]      = fmaf(y0, dd, bc0);
    A[row * HD + col + 16] = fmaf(y1, dd, bc1);
  }
}

// scalar tail (rows not covered by 32-row WMMA tiles; unused when N%32==0)
__global__ void __launch_bounds__(128)
k_gemm_tail(const float* __restrict__ Z, const float* __restrict__ W,
            float* __restrict__ Y, float* __restrict__ A,
            const float* __restrict__ dinv, const float* __restrict__ bias,
            int row0) {
  int n = row0 + blockIdx.x, c = threadIdx.x;
  __shared__ float zr[HD];
  zr[c] = Z[n * HD + c];
  __syncthreads();
  float t = 0.f;
  for (int k = 0; k < HD; ++k) t = fmaf(zr[k], W[k * HD + c], t);
  float dv = dinv[n];
  Y[n * HD + c] = t;
  A[n * HD + c] = fmaf(t, dv * dv, bias[c]);
}

// ---------------- small utility kernels -------------------------------------
__global__ void k_fill(float* p, float v, int n) {
  int i = blockIdx.x * blockDim.x + threadIdx.x;
  if (i < n) p[i] = v;
}

__global__ void k_deg(const int* __restrict__ dst, const float* __restrict__ ew,
                      float* __restrict__ deg, int E) {
  int e = blockIdx.x * blockDim.x + threadIdx.x;
  if (e < E) atomic_add_f32(&deg[dst[e]], ew[e]);
}

__global__ void k_rsqrt(float* p, int n) {
  int i = blockIdx.x * blockDim.x + threadIdx.x;
  if (i < n) p[i] = rsqrtf(p[i]);  // deg >= 1 (self loop), always > 0
}

__global__ void k_norm(const int* __restrict__ src, const int* __restrict__ dst,
                       const float* __restrict__ ew, const float* __restrict__ dinv,
                       float* __restrict__ nrm, int E) {
  int e = blockIdx.x * blockDim.x + threadIdx.x;
  if (e < E) nrm[e] = dinv[src[e]] * ew[e] * dinv[dst[e]];
}

// ---- edge scatter: one wave32 per edge, float4 per lane (512B gather) ------
__global__ void __launch_bounds__(256)
k_scatter(const float* __restrict__ hlin, const float* __restrict__ nrm,
          const int* __restrict__ src, const int* __restrict__ dst,
          float* __restrict__ acc, int E) {
  int e = (blockIdx.x * 256 + threadIdx.x) >> 5;  // global wave id == edge id
  int lane = threadIdx.x & 31;
  if (e >= E) return;
  // prefetch a future edge's source row into cache (global_prefetch_b8)
  if (e + 64 < E) __builtin_prefetch(hlin + (size_t)src[e + 64] * HD + lane * 4, 0, 1);
  int s = src[e], d = dst[e];
  float wt = nrm[e];
  float4 v = ((const float4*)(hlin + (size_t)s * HD))[lane];
  float* ad = acc + (size_t)d * HD + lane * 4;
  atomic_add_f32(ad + 0, v.x * wt);
  atomic_add_f32(ad + 1, v.y * wt);
  atomic_add_f32(ad + 2, v.z * wt);
  atomic_add_f32(ad + 3, v.w * wt);
}

// ---- pooling, fused with final residual: v = relu(zacc)+h ------------------
__global__ void __launch_bounds__(256)
k_pool(const float* __restrict__ zacc, const float* __restrict__ h,
       const int* __restrict__ batch, float* __restrict__ psum,
       unsigned* __restrict__ pmax, float* __restrict__ cnt, int N) {
  int n = (blockIdx.x * 256 + threadIdx.x) >> 5;  // one wave per node
  int lane = threadIdx.x & 31;
  if (n >= N) return;
  int g = batch[n];
  float4 a = ((const float4*)(zacc + (size_t)n * HD))[lane];
  float4 r = ((const float4*)(h + (size_t)n * HD))[lane];
  float v[4];
  v[0] = fmaxf(a.x, 0.0f) + r.x;
  v[1] = fmaxf(a.y, 0.0f) + r.y;
  v[2] = fmaxf(a.z, 0.0f) + r.z;
  v[3] = fmaxf(a.w, 0.0f) + r.w;
  float* ps = psum + (size_t)g * HD + lane * 4;
  unsigned* pm = pmax + (size_t)g * HD + lane * 4;
#pragma unroll
  for (int j = 0; j < 4; ++j) {
    atomic_add_f32(ps + j, v[j]);
    unsigned u = __float_as_uint(v[j]);
    u = (u & 0x80000000u) ? ~u : (u | 0x80000000u);  // order-preserving code
    atomicMax(pm + j, u);
  }
  if (lane == 0) atomic_add_f32(&cnt[g], 1.0f);
}

__global__ void __launch_bounds__(320)
k_feat(const float* __restrict__ psum, const unsigned* __restrict__ pmax,
       const float* __restrict__ cnt, const float* __restrict__ gf2,
       float* __restrict__ feat) {
  int g = blockIdx.x, t = threadIdx.x;
  float v;
  if (t < HD) {
    v = psum[g * HD + t] / fmaxf(cnt[g], 1.0f);
  } else if (t < 2 * HD) {
    unsigned u = pmax[g * HD + (t - HD)];
    if (u == 0u) v = 0.0f;  // empty segment -> reference maps -inf to 0
    else v = (u & 0x80000000u) ? __uint_as_float(u ^ 0x80000000u)
                               : __uint_as_float(~u);
  } else {
    v = gf2[g * 64 + (t - 2 * HD)];
  }
  feat[g * (2 * HD + 64) + t] = v;
}

// ---------------- generic tiny dense layer (G=128 rows) ---------------------
__global__ void k_dense(const float* __restrict__ in, const float* __restrict__ W,
                        const float* __restrict__ bias, float* __restrict__ out,
                        int K, int Nc, int relu) {
  int g = blockIdx.x, c = threadIdx.x;
  if (c >= Nc) return;
  float t = bias[c];
  for (int k = 0; k < K; ++k) t = fmaf(in[g * K + k], W[k * Nc + c], t);
  if (relu) t = fmaxf(t, 0.0f);
  out[g * Nc + c] = t;
}

// ---------------------------------------------------------------------------
extern "C" void kernel_launch(void* const* d_in, const int* in_sizes, int n_in,
                              void* d_out, int out_size, void* d_ws, size_t ws_size,
                              hipStream_t stream) {
  const float* x     = (const float*)d_in[0];
  const int*   ei    = (const int*)d_in[1];
  const float* ew    = (const float*)d_in[2];
  const int*   batch = (const int*)d_in[3];
  const float* gene  = (const float*)d_in[4];
  const float* W_in  = (const float*)d_in[5];
  const float* b_in  = (const float*)d_in[6];
  const float* lnig  = (const float*)d_in[7];
  const float* lnib  = (const float*)d_in[8];
  const float* lng   = (const float*)d_in[9];
  const float* lnb   = (const float*)d_in[10];
  const float* convW = (const float*)d_in[11];
  const float* convb = (const float*)d_in[12];
  const float* Wg1   = (const float*)d_in[13];
  const float* bg1   = (const float*)d_in[14];
  const float* Wg2   = (const float*)d_in[15];
  const float* bg2   = (const float*)d_in[16];
  const float* Wm1   = (const float*)d_in[17];
  const float* bm1   = (const float*)d_in[18];
  const float* Wm2   = (const float*)d_in[19];
  const float* bm2   = (const float*)d_in[20];
  const float* Wm3   = (const float*)d_in[21];
  const float* bm3   = (const float*)d_in[22];

  const int E = in_sizes[2];
  const int N = in_sizes[3];
  const int G = in_sizes[4] / 66;
  const int L = in_sizes[9] / HD;
  const int* src = ei;
  const int* dst = ei + E;

  // workspace partition (~157 MB)
  float*    h    = (float*)d_ws;
  float*    z    = h + (size_t)N * HD;      // doubles as scatter accumulator
  float*    hlin = z + (size_t)N * HD;
  float*    dinv = hlin + (size_t)N * HD;
  float*    nrm  = dinv + N;
  float*    psum = nrm + E;
  unsigned* pmax = (unsigned*)(psum + (size_t)G * HD);
  float*    cnt  = (float*)(pmax + (size_t)G * HD);
  float*    gf1  = cnt + G;
  float*    gf2  = gf1 + (size_t)G * HD;
  float*    feat = gf2 + (size_t)G * 64;
  float*    m1   = feat + (size_t)G * (2 * HD + 64);
  float*    m2   = m1 + (size_t)G * HD;

  // degrees (layer-invariant): deg = 1 (self loop) + sum of incoming ew
  k_fill<<<(N + 255) / 256, 256, 0, stream>>>(dinv, 1.0f, N);
  k_deg<<<(E + 255) / 256, 256, 0, stream>>>(dst, ew, dinv, E);
  k_rsqrt<<<(N + 255) / 256, 256, 0, stream>>>(dinv, N);
  k_norm<<<(E + 255) / 256, 256, 0, stream>>>(src, dst, ew, dinv, nrm, E);

  // input projection, then first LayerNorm
  k_inproj<<<N, 128, 0, stream>>>(x, W_in, b_in, lnig, lnib, h);
  k_ln<<<N, 128, 0, stream>>>(h, z, lng, lnb);

  // residual GCN layers
  for (int i = 0; i < L; ++i) {
    const float* Wi = convW + (size_t)i * HD * HD;
    const float* bi = convb + (size_t)i * HD;
    int full = N / 32;
    if (full) k_gemm<<<full, 256, 0, stream>>>(z, Wi, hlin, z, dinv, bi);
    int rem = N - full * 32;
    if (rem) k_gemm_tail<<<rem, 128, 0, stream>>>(z, Wi, hlin, z, dinv, bi, full * 32);
    k_scatter<<<(E + 7) / 8, 256, 0, stream>>>(hlin, nrm, src, dst, z, E);
    if (i + 1 < L)  // fuse residual+relu with next layer's LayerNorm
      k_resid_ln<<<N, 128, 0, stream>>>(z, h, lng + (i + 1) * HD, lnb + (i + 1) * HD);
  }

  // pooling (fused with last layer's residual+relu)
  k_fill<<<(G * HD + 255) / 256, 256, 0, stream>>>(psum, 0.0f, G * HD);
  k_fill<<<(G * HD + 255) / 256, 256, 0, stream>>>((float*)pmax, 0.0f, G * HD);
  k_fill<<<(G + 255) / 256, 256, 0, stream>>>(cnt, 0.0f, G);
  k_pool<<<(N + 7) / 8, 256, 0, stream>>>(z, h, batch, psum, pmax, cnt, N);

  // gene feature projection
  k_dense<<<G, 128, 0, stream>>>(gene, Wg1, bg1, gf1, 66, HD, 1);
  k_dense<<<G, 64, 0, stream>>>(gf1, Wg2, bg2, gf2, HD, 64, 1);

  // concat features + classifier MLP
  k_feat<<<G, 320, 0, stream>>>(psum, pmax, cnt, gf2, feat);
  k_dense<<<G, 128, 0, stream>>>(feat, Wm1, bm1, m1, 2 * HD + 64, HD, 1);
  k_dense<<<G, 64, 0, stream>>>(m1, Wm2, bm2, m2, HD, 64, 1);
  k_dense<<<G, 32, 0, stream>>>(m2, Wm3, bm3, (float*)d_out, 64, 2, 0);
}